// CTCLayer_2894807957577
// MI455X (gfx1250) — compile-verified
//
#include <hip/hip_runtime.h>
#include <cstdint>

#define BT 256   // time steps
#define BC 128   // classes
#define BL 32    // label length
#define BS 65    // 2*BL+1 extended states
#define CHUNK 4  // rows per async stage (one row per wave)
#define NEGV (-1e30f)

__device__ __forceinline__ float wmaxf(float v) {
#pragma unroll
  for (int m = 16; m; m >>= 1) v = fmaxf(v, __shfl_xor(v, m, 32));
  return v;
}
__device__ __forceinline__ float wsumf(float v) {
#pragma unroll
  for (int m = 16; m; m >>= 1) v += __shfl_xor(v, m, 32);
  return v;
}

__global__ __launch_bounds__(128) void ctc_loss_kernel(
    const int* __restrict__ y_true,   // [B, BL]
    const float* __restrict__ y_pred, // [B, BT, BC]
    float* __restrict__ out)          // [B]
{
  const int b    = blockIdx.x;
  const int tid  = threadIdx.x;
  const int lane = tid & 31;
  const int wid  = tid >> 5;

  // 16B alignment required: async b128 writes 16B per lane into LDS.
  __shared__ alignas(16) float xbuf[2][CHUNK * BC]; // double-buffered logit rows (2 x 2KB)
  __shared__ float alpha[2][BS + 2];                // ping-pong alphas, 2-slot NEG pad
  __shared__ float pmax[4], psum[4];

  // Per-state extended label + skip-transition legality (states are thread ids < BS)
  int extS = 0, canSkip = 0;
  if (tid < BS) {
    if (tid & 1) {
      const int li = tid >> 1;
      extS = y_true[b * BL + li];                       // labels in [1, C-1]
      if (tid >= 2) canSkip = (extS != y_true[b * BL + li - 1]);
    } // even states: blank (= index 0), canSkip stays 0
  }
  if (tid < 2) { alpha[0][tid] = NEGV; alpha[1][tid] = NEGV; }

  const float* ybase = y_pred + (size_t)b * BT * BC;

  // Prologue: async-DMA chunk 0 into stage 0. Each wave loads one row:
  // 32 lanes x b128 (16B) = 512B = one [BC] f32 row. Non-temporal: 268MB
  // one-shot stream, keep it out of L2 retention.
  {
    uint64_t g = (uint64_t)(uintptr_t)(ybase + (size_t)wid * BC) + (uint64_t)(lane * 16);
    uint32_t l = (uint32_t)(uintptr_t)(&xbuf[0][wid * BC]) + (uint32_t)(lane * 16);
    asm volatile("global_load_async_to_lds_b128 %0, %1, off th:TH_LOAD_NT"
                 :: "v"(l), "v"(g) : "memory");
  }

  const int NC = BT / CHUNK;
  for (int c = 0; c < NC; ++c) {
    // Kick off next chunk's DMA, then wait for the current chunk (in-order completion).
    if (c + 1 < NC) {
      const int row = (c + 1) * CHUNK + wid;
      uint64_t g = (uint64_t)(uintptr_t)(ybase + (size_t)row * BC) + (uint64_t)(lane * 16);
      uint32_t l = (uint32_t)(uintptr_t)(&xbuf[(c + 1) & 1][wid * BC]) + (uint32_t)(lane * 16);
      asm volatile("global_load_async_to_lds_b128 %0, %1, off th:TH_LOAD_NT"
                   :: "v"(l), "v"(g) : "memory");
      asm volatile("s_wait_asynccnt 0x1" ::: "memory");
    } else {
      asm volatile("s_wait_asynccnt 0x0" ::: "memory");
    }
    __syncthreads();

    const float* stagebuf = xbuf[c & 1];
#pragma unroll
    for (int r = 0; r < CHUNK; ++r) {
      const int t = c * CHUNK + r;
      const float* x = stagebuf + r * BC;

      // --- stable logsumexp over C=128 (wave32 shuffles + 4-partial combine) ---
      const float v  = x[tid];
      const float wm = wmaxf(v);
      if (lane == 0) pmax[wid] = wm;
      __syncthreads();
      const float m  = fmaxf(fmaxf(pmax[0], pmax[1]), fmaxf(pmax[2], pmax[3]));
      const float ws = wsumf(expf(v - m));
      if (lane == 0) psum[wid] = ws;
      __syncthreads();
      const float lse = m + logf(psum[0] + psum[1] + psum[2] + psum[3]);

      // --- CTC alpha recursion in log space ---
      if (tid < BS) {
        const float xe = x[extS] - lse;   // log-softmax gathered at extended label
        float anew;
        if (t == 0) {
          anew = (tid < 2) ? xe : NEGV;
        } else {
          const float* ap = alpha[(t - 1) & 1];
          const float a0 = ap[2 + tid];
          const float a1 = ap[1 + tid];
          const float a2 = canSkip ? ap[tid] : NEGV;
          const float mm = fmaxf(a0, fmaxf(a1, a2));
          // exp of <= 0 args; masked/NEG terms underflow to exactly 0 — NaN-free
          anew = mm + logf(expf(a0 - mm) + expf(a1 - mm) + expf(a2 - mm)) + xe;
        }
        alpha[t & 1][2 + tid] = anew;
      }
      __syncthreads();
    }
  }

  if (tid == 0) {
    const float* af = alpha[(BT - 1) & 1];
    const float a  = af[2 + BS - 1];
    const float bb = af[2 + BS - 2];
    const float mx = fmaxf(a, bb);
    const float mn = fminf(a, bb);
    out[b] = -(mx + log1pf(expf(mn - mx)));
  }
}

extern "C" void kernel_launch(void* const* d_in, const int* in_sizes, int n_in,
                              void* d_out, int out_size, void* d_ws, size_t ws_size,
                              hipStream_t stream) {
  (void)n_in; (void)d_ws; (void)ws_size; (void)out_size;
  const int*   y_true = (const int*)d_in[0];    // [B, BL] int32
  const float* y_pred = (const float*)d_in[1];  // [B, BT, BC] f32
  float*       out    = (float*)d_out;          // [B] f32
  const int B = in_sizes[0] / BL;               // 2048
  ctc_loss_kernel<<<dim3(B), dim3(128), 0, stream>>>(y_true, y_pred, out);
}